// CPDBlock_76794015252790
// MI455X (gfx1250) — compile-verified
//
#include <hip/hip_runtime.h>

// CPD block: 1x1 conv (64->64, fp32 WMMA GEMM) -> 1x3 dw conv -> 3x1 dw conv + bias.
// MI455X plan: two passes, 134MB intermediate y kept hot in 192MB L2.
//   Pass 1: y = W1 * X via V_WMMA_F32_16X16X4_F32. X tiles staged into LDS by the
//           Tensor Data Mover (TENSOR_LOAD_TO_LDS, TENSORcnt), double-buffered,
//           with TDM pad_interval/pad_amount producing the bank-conflict-free
//           72-dword row stride in hardware. Async-LDS fallback if no TDM builtin.
//   Pass 2: fused separable 3x3 depthwise + bias from L2-resident y.

typedef __attribute__((ext_vector_type(2))) float        v2f;
typedef __attribute__((ext_vector_type(8))) float        v8f;
typedef __attribute__((ext_vector_type(4))) int          v4i;
typedef __attribute__((ext_vector_type(4))) unsigned int u32x4;
typedef __attribute__((ext_vector_type(8))) int          i32x8;
typedef __attribute__((ext_vector_type(4))) int          i32x4;

#define IN_C   64
#define OUT_C  64
#define Hdim   256
#define Wdim   256
#define HW     (Hdim * Wdim)
#define PIX_TILE 64          // pixels per tile (4 N-tiles of 16)
#define TILES_PER_BLK 4      // 256 contiguous pixels per block, double-buffered
#define XS_STRIDE 72         // LDS row stride (floats): 64 data + 8 TDM pad dwords

// ---- CDNA5 TDM staging (TENSORcnt), with async-LDS (ASYNCcnt) fallback ----
#if __has_builtin(__builtin_amdgcn_tensor_load_to_lds)
  #define HAVE_TDM 1
  #if __has_include(<hip/amd_detail/amd_gfx1250_TDM.h>)
    // amdgpu-toolchain (therock): 6-arg form
    #define TDM_LOAD(g0, g1) \
        __builtin_amdgcn_tensor_load_to_lds((g0), (g1), (i32x4){0,0,0,0}, \
            (i32x4){0,0,0,0}, (i32x8){0,0,0,0,0,0,0,0}, 0)
  #else
    // ROCm 7.2 (clang-22): 5-arg form
    #define TDM_LOAD(g0, g1) \
        __builtin_amdgcn_tensor_load_to_lds((g0), (g1), (i32x4){0,0,0,0}, \
            (i32x4){0,0,0,0}, 0)
  #endif
#endif

#if __has_builtin(__builtin_amdgcn_global_load_async_to_lds_b128)
  #define ASYNC_CP16(srcp, dstp)                                              \
      __builtin_amdgcn_global_load_async_to_lds_b128(                         \
          (v4i*)(srcp), (v4i*)(dstp), 0, 0)
#else
  #define ASYNC_CP16(srcp, dstp)                                              \
      do {                                                                    \
          unsigned _lds = (unsigned)(unsigned long long)                      \
              (__attribute__((address_space(3))) float*)(dstp);               \
          unsigned long long _ga = (unsigned long long)(srcp);                \
          asm volatile("global_load_async_to_lds_b128 %0, %1, off"            \
                       :: "v"(_lds), "v"(_ga) : "memory");                    \
      } while (0)
#endif

#if __has_builtin(__builtin_amdgcn_s_wait_asynccnt)
  #define WAIT_ASYNC(nlit) __builtin_amdgcn_s_wait_asynccnt(nlit)
#else
  #define WAIT_ASYNC(nlit) asm volatile("s_wait_asynccnt %0" :: "i"(nlit) : "memory")
#endif

// ---------------------------------------------------------------------------
// Pass 1: y[b,o,p] = sum_c w1[o,c] * x[b,c,p]     (pure GEMM on the matrix pipe)
// ---------------------------------------------------------------------------
__global__ __launch_bounds__(128)
void cpd_proj_wmma(const float* __restrict__ x,
                   const float* __restrict__ w1,
                   float* __restrict__ y)
{
    __shared__ float xs[2][IN_C * XS_STRIDE];   // 2 x 18 KB double buffer

    const int tid  = threadIdx.x;
    const int wave = tid >> 5;               // 4 waves: each owns 16 output channels
    const int lane = tid & 31;
    const int hi   = lane >> 4;              // half-wave select (WMMA layout)
    const int lm   = lane & 15;

    const int gid   = blockIdx.x;            // 2048 blocks: [b][256-px span]
    const int b     = gid >> 8;              // 256 spans per batch
    const int pbase = (gid & 255) * (PIX_TILE * TILES_PER_BLK);

    const float* xb = x + (long long)b * IN_C * HW;
    float*       yb = y + (long long)b * OUT_C * HW;

    // Preload this wave's A fragments once: W1 rows [m0, m0+16), all K.
    // fp32 A 16x4 layout: lanes 0-15 = M, VGPR0 K={0 | 2}, VGPR1 K={1 | 3}.
    const int m0 = wave * 16;
    v2f afrag[16];
#pragma unroll
    for (int kb = 0; kb < 16; ++kb) {
        const float* ap = w1 + (m0 + lm) * IN_C + kb * 4 + 2 * hi;
        afrag[kb] = *(const v2f*)ap;         // K = {2*hi, 2*hi+1}
    }

#if defined(HAVE_TDM)
    // ---- TDM descriptor-driven staging: one instruction per 16 KB tile ----
    // D# group1 (constant): data_size=4B, pad every 64 dwords by 8 dwords
    // (rows land at XS_STRIDE=72), tile 64px x 64ch, tensor dim0=HW, dim1=64,
    // dim0_stride=HW elements.
    i32x8 g1;
    g1[0] = (2 << 16) | (1 << 20) | (5 << 22) | (7 << 25);
    g1[1] = 0;                                   // atomic-barrier addr unused
    g1[2] = (int)(((unsigned)HW >> 16) | ((unsigned)IN_C << 16)); // dim0 hi | dim1 lo
    g1[3] = (int)((unsigned)PIX_TILE << 16);     // dim1 hi(0) | tile_dim0
    g1[4] = IN_C;                                // tile_dim1 | tile_dim2(0)
    g1[5] = HW;                                  // dim0_stride low32
    g1[6] = 0;                                   // dim0_stride hi | dim1_stride lo
    g1[7] = 0;

    auto issue_tile = [&](int t, int bi) {
        const unsigned long long ga =
            (unsigned long long)(xb + pbase + t * PIX_TILE);
        const unsigned lds_addr = (unsigned)(unsigned long long)
            (__attribute__((address_space(3))) float*)(&xs[bi][0]);
        u32x4 g0;
        g0[0] = 1u;                              // count=1, user descriptor
        g0[1] = lds_addr;                        // LDS byte address
        g0[2] = (unsigned)(ga & 0xFFFFFFFFu);    // global_addr[31:0]
        g0[3] = (unsigned)((ga >> 32) & 0x01FFFFFFu) | 0x80000000u; // type=2
        TDM_LOAD(g0, g1);
    };

    if (wave == 0) issue_tile(0, 0);
#else
    // ---- fallback: per-thread async global->LDS b128 copies ----
    auto issue_tile = [&](int t, int bi) {
#pragma unroll
        for (int i = 0; i < 8; ++i) {
            const int d4  = i * 128 + tid;       // 1024 float4 per tile
            const int c   = d4 >> 4;
            const int px4 = (d4 & 15) << 2;
            const float* src = xb + (long long)c * HW + pbase + t * PIX_TILE + px4;
            ASYNC_CP16(src, &xs[bi][c * XS_STRIDE + px4]);
        }
    };
    issue_tile(0, 0);
#endif

    for (int t = 0; t < TILES_PER_BLK; ++t) {
        const int bi = t & 1;
#if defined(HAVE_TDM)
        if (wave == 0) {
            if (t + 1 < TILES_PER_BLK) {
                issue_tile(t + 1, (t + 1) & 1);
                __builtin_amdgcn_s_wait_tensorcnt(1);  // tile t done (in-order)
            } else {
                __builtin_amdgcn_s_wait_tensorcnt(0);
            }
        }
#else
        if (t + 1 < TILES_PER_BLK) {
            issue_tile(t + 1, (t + 1) & 1);
            WAIT_ASYNC(8);                       // tile t's 8 copies complete
        } else {
            WAIT_ASYNC(0);
        }
#endif
        __syncthreads();                         // tile t visible to all waves

        const int p0 = pbase + t * PIX_TILE;
        const float* xsb = xs[bi];

        // 4 N-tiles of 16 pixels; K = 64 in 16 steps of 4
#pragma unroll
        for (int nt = 0; nt < 4; ++nt) {
            const int n = nt * 16 + lm;
            v8f acc = {};
#pragma unroll
            for (int kb = 0; kb < 16; ++kb) {
                const int c = kb * 4 + 2 * hi;
                v2f bfrag;
                bfrag.x = xsb[(c + 0) * XS_STRIDE + n];
                bfrag.y = xsb[(c + 1) * XS_STRIDE + n];
                // 8 args: (neg_a, A, neg_b, B, c_mod, C, reuse_a, reuse_b)
                acc = __builtin_amdgcn_wmma_f32_16x16x4_f32(
                    false, afrag[kb], false, bfrag, (short)0, acc, false, false);
            }
            // D layout: VGPR r, lanes<16 -> M=r, lanes>=16 -> M=r+8; N = lane&15
#pragma unroll
            for (int r = 0; r < 8; ++r) {
                yb[(long long)(m0 + r + 8 * hi) * HW + p0 + nt * 16 + lm] = acc[r];
            }
        }
        __syncthreads();                         // done reading xs[bi] before reuse
    }
}

// ---------------------------------------------------------------------------
// Pass 2: out = conv_v(conv_h(y)) + bias, separable 3x3 taps, y from L2
// ---------------------------------------------------------------------------
#define HT 4
#define WT 64

__global__ __launch_bounds__(256)
void cpd_dwconv(const float* __restrict__ y,
                const float* __restrict__ wh,
                const float* __restrict__ wv,
                const float* __restrict__ bias,
                float* __restrict__ out)
{
    __shared__ float ys[6][WT + 4];          // rows h0-1..h0+4, cols w0-1..w0+64

    const int tid = threadIdx.x;
    int gid = blockIdx.x;                    // [b][o][hTile][wTile] = 8*64*64*4
    const int wTile = gid & 3;   gid >>= 2;
    const int hTile = gid & 63;  gid >>= 6;
    const int o     = gid & 63;  gid >>= 6;
    const int b     = gid;

    const int h0 = hTile * HT;
    const int w0 = wTile * WT;

    const float* yc = y + ((long long)b * OUT_C + o) * HW;

    // halo window load with zero padding at image borders
    for (int i = tid; i < 6 * 66; i += 256) {
        const int r  = i / 66;
        const int cc = i - r * 66;
        const int gh = h0 + r - 1;
        const int gw = w0 + cc - 1;
        float v = 0.0f;
        if ((unsigned)gh < (unsigned)Hdim && (unsigned)gw < (unsigned)Wdim)
            v = yc[gh * Wdim + gw];
        ys[r][cc] = v;
    }

    // per-channel taps (uniform -> scalar loads)
    const float wh0 = wh[o * 3 + 0], wh1 = wh[o * 3 + 1], wh2 = wh[o * 3 + 2];
    const float wv0 = wv[o * 3 + 0], wv1 = wv[o * 3 + 1], wv2 = wv[o * 3 + 2];
    const float bo  = bias[o];

    __syncthreads();

    const int r  = tid >> 6;                 // 0..3 output row in tile
    const int cw = tid & 63;                 // 0..63 output col in tile

    float acc = bo;
    {
        float row0 = ys[r + 0][cw] * wh0 + ys[r + 0][cw + 1] * wh1 + ys[r + 0][cw + 2] * wh2;
        float row1 = ys[r + 1][cw] * wh0 + ys[r + 1][cw + 1] * wh1 + ys[r + 1][cw + 2] * wh2;
        float row2 = ys[r + 2][cw] * wh0 + ys[r + 2][cw + 1] * wh1 + ys[r + 2][cw + 2] * wh2;
        acc = fmaf(wv0, row0, acc);
        acc = fmaf(wv1, row1, acc);
        acc = fmaf(wv2, row2, acc);
    }
    out[(((long long)b * OUT_C + o) * Hdim + (h0 + r)) * Wdim + w0 + cw] = acc;
}

// ---------------------------------------------------------------------------
extern "C" void kernel_launch(void* const* d_in, const int* in_sizes, int n_in,
                              void* d_out, int out_size, void* d_ws, size_t ws_size,
                              hipStream_t stream)
{
    const float* x    = (const float*)d_in[0];   // [8,64,256,256]
    const float* w1   = (const float*)d_in[1];   // [64,64]
    const float* wh   = (const float*)d_in[2];   // [64,3]
    const float* wv   = (const float*)d_in[3];   // [64,3]
    const float* bias = (const float*)d_in[4];   // [64]
    float* out = (float*)d_out;                  // [8,64,256,256]
    float* y   = (float*)d_ws;                   // 128 MiB intermediate, L2-resident

    // Pass 1: WMMA GEMM (1x1 conv). 8 batches * 256 spans of 256 px, 128 thr (4 waves).
    cpd_proj_wmma<<<8 * (HW / (PIX_TILE * TILES_PER_BLK)), 128, 0, stream>>>(x, w1, y);

    // Pass 2: separable depthwise 3x3 + bias. 8*64*(256/4)*(256/64) blocks, 256 thr.
    cpd_dwconv<<<8 * OUT_C * (Hdim / HT) * (Wdim / WT), 256, 0, stream>>>(y, wh, wv, bias, out);
}